// AttentionPooler_27513560498678
// MI455X (gfx1250) — compile-verified
//
#include <hip/hip_runtime.h>
#include <hip/hip_bf16.h>

typedef float v2f __attribute__((ext_vector_type(2)));
typedef float v8f __attribute__((ext_vector_type(8)));

#define DIM            1024
#define NBATCH         8
#define SEQ            8192
#define ROWS_PER_BLOCK 128
#define CHUNKS         (SEQ / ROWS_PER_BLOCK)     // 64
#define NPART          (NBATCH * CHUNKS)          // 512

// ---------------------------------------------------------------------------
// Kernel 1: per-chunk flash partial.
// Each block: 256 threads (8 waves), owns 128 rows of one batch.
//   - scores via V_WMMA_F32_16X16X4_F32 (A = broadcast weight, B = feat rows)
//   - local max m, local l = sum exp(s - m)
//   - acc[d] = sum_n exp(s_n - m) * feats[n, d]   (coalesced float4 pass)
// ---------------------------------------------------------------------------
__global__ __launch_bounds__(256) void pool_partial_kernel(
    const float* __restrict__ feats, const float* __restrict__ weight,
    float* __restrict__ pm, float* __restrict__ pl, float* __restrict__ pacc)
{
    __shared__ float w_lds[DIM];
    __shared__ float s_lds[ROWS_PER_BLOCK];
    __shared__ float p_lds[ROWS_PER_BLOCK];

    const int t     = threadIdx.x;
    const int b     = blockIdx.x / CHUNKS;
    const int chunk = blockIdx.x % CHUNKS;
    const long row0 = (long)b * SEQ + (long)chunk * ROWS_PER_BLOCK;
    const float* __restrict__ fbase = feats + row0 * DIM;

    // Stage weight vector (4 KB) into LDS, coalesced float4.
    ((float4*)w_lds)[t] = ((const float4*)weight)[t];
    __syncthreads();

    const int wave = t >> 5;
    const int lane = t & 31;
    const int n    = lane & 15;   // row within 16-row tile
    const int hi   = lane >> 4;   // selects K pair {0,1} vs {2,3}
    const int tile_row = wave * 16;
    const float* __restrict__ frow = fbase + (long)(tile_row + n) * DIM;

    // C[m,n] = sum_k w[k] * feats[row n, k]  -> score[n] replicated over M.
    v8f c = {0.f, 0.f, 0.f, 0.f, 0.f, 0.f, 0.f, 0.f};
#pragma unroll 8
    for (int k0 = 0; k0 < DIM; k0 += 4) {
        const int ka = k0 + 2 * hi;           // even -> 8B aligned
        v2f a;
        a.x = w_lds[ka];
        a.y = w_lds[ka + 1];
        v2f bb = *(const v2f*)(frow + ka);
        // 8 args: (neg_a, A, neg_b, B, c_mod, C, reuse_a, reuse_b)
        c = __builtin_amdgcn_wmma_f32_16x16x4_f32(
                false, a, false, bb, (short)0, c, false, false);
    }

    // Every lane L holds score of row (tile_row + L%16) in c[0].
    if (lane < 16) s_lds[tile_row + n] = c[0];
    __syncthreads();

    // Block-wide max over 128 scores (每 thread redundantly; LDS broadcast reads).
    float m = -INFINITY;
#pragma unroll 8
    for (int i = 0; i < ROWS_PER_BLOCK; ++i) m = fmaxf(m, s_lds[i]);

    if (t < ROWS_PER_BLOCK) p_lds[t] = __expf(s_lds[t] - m);
    __syncthreads();

    float l = 0.f;
#pragma unroll 8
    for (int i = 0; i < ROWS_PER_BLOCK; ++i) l += p_lds[i];

    if (t == 0) { pm[blockIdx.x] = m; pl[blockIdx.x] = l; }

    // Weighted accumulation: thread t owns d = 4t..4t+3 (coalesced 4 KB rows,
    // chunk should still be resident in L2 from the score pass).
    const int d0 = t * 4;
    float ax = 0.f, ay = 0.f, az = 0.f, aw = 0.f;
#pragma unroll 4
    for (int r = 0; r < ROWS_PER_BLOCK; ++r) {
        const float p = p_lds[r];
        const float4 f = *(const float4*)(fbase + (long)r * DIM + d0);
        ax = fmaf(p, f.x, ax);
        ay = fmaf(p, f.y, ay);
        az = fmaf(p, f.z, az);
        aw = fmaf(p, f.w, aw);
    }
    float4 outv; outv.x = ax; outv.y = ay; outv.z = az; outv.w = aw;
    *(float4*)(pacc + (long)blockIdx.x * DIM + d0) = outv;
}

// ---------------------------------------------------------------------------
// Kernel 2: combine 64 chunk-partials per batch.
// out[b,d] = ( sum_i exp(m_i - M) * acc_i[d] ) / ( sum_i l_i * exp(m_i - M) )
// ---------------------------------------------------------------------------
__global__ __launch_bounds__(256) void pool_combine_kernel(
    const float* __restrict__ pm, const float* __restrict__ pl,
    const float* __restrict__ pacc, float* __restrict__ out)
{
    const int b = blockIdx.x;
    const int t = threadIdx.x;
    const float* m = pm + b * CHUNKS;
    const float* l = pl + b * CHUNKS;

    float M = -INFINITY;
#pragma unroll 8
    for (int i = 0; i < CHUNKS; ++i) M = fmaxf(M, m[i]);
    float Z = 0.f;
#pragma unroll 8
    for (int i = 0; i < CHUNKS; ++i) Z += l[i] * __expf(m[i] - M);
    const float invZ = 1.f / Z;

    const int d0 = t * 4;
    float ax = 0.f, ay = 0.f, az = 0.f, aw = 0.f;
    for (int i = 0; i < CHUNKS; ++i) {
        const float s = __expf(m[i] - M);
        const float4 a =
            *(const float4*)(pacc + ((long)(b * CHUNKS + i)) * DIM + d0);
        ax = fmaf(s, a.x, ax);
        ay = fmaf(s, a.y, ay);
        az = fmaf(s, a.z, az);
        aw = fmaf(s, a.w, aw);
    }
    float4 o;
    o.x = ax * invZ; o.y = ay * invZ; o.z = az * invZ; o.w = aw * invZ;
    *(float4*)(out + (long)b * DIM + d0) = o;
}

// ---------------------------------------------------------------------------
extern "C" void kernel_launch(void* const* d_in, const int* in_sizes, int n_in,
                              void* d_out, int out_size, void* d_ws, size_t ws_size,
                              hipStream_t stream)
{
    (void)in_sizes; (void)n_in; (void)out_size; (void)ws_size;
    const float* feats  = (const float*)d_in[0];   // [8, 8192, 1024] f32
    const float* weight = (const float*)d_in[1];   // [1024] f32
    float* out = (float*)d_out;                    // [8, 1024] f32

    // Workspace layout: pm[512] | pl[512] | pacc[512*1024] (16B-aligned blocks)
    char* ws = (char*)d_ws;
    float* pm   = (float*)(ws);
    float* pl   = (float*)(ws + 2048);
    float* pacc = (float*)(ws + 4096);

    pool_partial_kernel<<<NPART, 256, 0, stream>>>(feats, weight, pm, pl, pacc);
    pool_combine_kernel<<<NBATCH, 256, 0, stream>>>(pm, pl, pacc, out);
}